// HeteroGNNLayer_39376260169906
// MI455X (gfx1250) — compile-verified
//
#include <hip/hip_runtime.h>
#include <math.h>

typedef __attribute__((ext_vector_type(2))) float v2f;
typedef __attribute__((ext_vector_type(8))) float v8f;

static constexpr int DIM = 128;   // feature dim
static constexpr int HEADS = 4;   // GAT heads

// ---------- helpers: monotonic float<->int mapping for atomicMax on floats ----------
__device__ __forceinline__ int ord_f32(float f) {
  int i = __float_as_int(f);
  return i >= 0 ? i : (i ^ 0x7FFFFFFF);
}
__device__ __forceinline__ float unord_f32(int i) {
  return __int_as_float(i >= 0 ? i : (i ^ 0x7FFFFFFF));
}

// ---------- fills ----------
__global__ void fill_f32(float* __restrict__ p, float v, size_t n) {
  size_t i = (size_t)blockIdx.x * blockDim.x + threadIdx.x;
  size_t st = (size_t)gridDim.x * blockDim.x;
  for (; i < n; i += st) p[i] = v;
}
__global__ void fill_i32(int* __restrict__ p, int v, size_t n) {
  size_t i = (size_t)blockIdx.x * blockDim.x + threadIdx.x;
  size_t st = (size_t)gridDim.x * blockDim.x;
  for (; i < n; i += st) p[i] = v;
}

// ---------- WMMA f32 GEMM, K = 128 fixed, one wave -> one 16x64 strip ----------
// out[M,N] = A1[M,128] @ W1[128,N] (+ A2[M,128] @ W2[128,N]) (+ bias[N])
// Requires M % 16 == 0, N % 64 == 0, blockDim.x == 32.
// Per k-step: 1 A-fragment load + 4 independent B-fragment loads (distinct regs),
// then 4 WMMAs -> loads overlap with matrix math instead of wait-0 serialization.
__device__ __forceinline__ void gemm_strip_k128(const float* __restrict__ A,
                                                const float* __restrict__ W,
                                                int tm, int tn0, int r, int kp, int N,
                                                v8f acc[4]) {
  const float* arow = A + (size_t)(tm * 16 + r) * DIM;
  const float* bcol = W + (size_t)tn0 + r;
  for (int k = 0; k < DIM; k += 4) {
    const float* b0 = bcol + (size_t)(k + kp) * N;
    const float* b1 = bcol + (size_t)(k + kp + 1) * N;
    v2f a, q0, q1, q2, q3;
    a.x = arow[k + kp];
    a.y = arow[k + kp + 1];
    q0.x = b0[0];  q0.y = b1[0];
    q1.x = b0[16]; q1.y = b1[16];
    q2.x = b0[32]; q2.y = b1[32];
    q3.x = b0[48]; q3.y = b1[48];
    acc[0] = __builtin_amdgcn_wmma_f32_16x16x4_f32(false, a, false, q0, (short)0, acc[0], false, false);
    acc[1] = __builtin_amdgcn_wmma_f32_16x16x4_f32(false, a, false, q1, (short)0, acc[1], false, false);
    acc[2] = __builtin_amdgcn_wmma_f32_16x16x4_f32(false, a, false, q2, (short)0, acc[2], false, false);
    acc[3] = __builtin_amdgcn_wmma_f32_16x16x4_f32(false, a, false, q3, (short)0, acc[3], false, false);
  }
}

__global__ void gemm_wmma_k128(const float* __restrict__ A1, const float* __restrict__ W1,
                               const float* __restrict__ A2, const float* __restrict__ W2,
                               const float* __restrict__ bias,
                               float* __restrict__ out, int N) {
  const int lane = threadIdx.x & 31;
  const int tn0 = blockIdx.x * 64;  // 4 consecutive 16-wide N tiles
  const int tm = blockIdx.y;
  const int half = lane >> 4;       // 0: K-pair {0,1}, 1: K-pair {2,3}
  const int r = lane & 15;          // A row / B col within tile
  const int kp = half << 1;

  v8f acc[4] = {{}, {}, {}, {}};
  gemm_strip_k128(A1, W1, tm, tn0, r, kp, N, acc);
  if (A2) gemm_strip_k128(A2, W2, tm, tn0, r, kp, N, acc);

  // C/D layout: VGPR j -> row (j + half*8), col = lane&15 within each 16-wide tile
  const int mbase = tm * 16 + (half << 3);
#pragma unroll
  for (int q = 0; q < 4; ++q) {
    const int col = tn0 + q * 16 + r;
    const float bv = bias ? bias[col] : 0.0f;
#pragma unroll
    for (int j = 0; j < 8; ++j)
      out[(size_t)(mbase + j) * N + col] = acc[q][j] + bv;
  }
}

// ---------- GAT: per-(node,head) attention logits ----------
// al_s[n,h] = dot(xp[n,h,:], att_src[h,:]); al_d likewise. i indexes n*H+h.
__global__ void att_logits(const float* __restrict__ xp, const float* __restrict__ a_src,
                           const float* __restrict__ a_dst,
                           float* __restrict__ al_s, float* __restrict__ al_d, int NH) {
  int i = blockIdx.x * blockDim.x + threadIdx.x;
  if (i >= NH) return;
  int h = i & (HEADS - 1);
  const float* row = xp + (size_t)i * DIM;   // n*H*D + h*D == i*D
  const float* as = a_src + h * DIM;
  const float* ad = a_dst + h * DIM;
  float ss = 0.f, sd = 0.f;
  for (int c = 0; c < DIM; ++c) {
    float v = row[c];
    ss += v * as[c];
    sd += v * ad[c];
  }
  al_s[i] = ss;
  al_d[i] = sd;
}

// ---------- GAT pass 1: segment max of leaky-relu logits per (dst, head) ----------
__global__ void gat_edge_max(const int* __restrict__ esrc, const int* __restrict__ edst,
                             const float* __restrict__ al_s, const float* __restrict__ al_d,
                             int* __restrict__ m_ord, int E) {
  int e = blockIdx.x * blockDim.x + threadIdx.x;
  if (e >= E) return;
  int sN = esrc[e], dN = edst[e];
#pragma unroll
  for (int h = 0; h < HEADS; ++h) {
    float l = al_s[sN * HEADS + h] + al_d[dN * HEADS + h];
    l = l > 0.f ? l : 0.2f * l;
    atomicMax(&m_ord[dN * HEADS + h], ord_f32(l));
  }
}

// ---------- GAT pass 2: unnormalized softmax accumulate ----------
// blockDim = 128: one block per edge, thread t -> head t>>5, lane t&31 covers 4 channels.
__global__ void gat_edge_acc(const int* __restrict__ esrc, const int* __restrict__ edst,
                             const float* __restrict__ al_s, const float* __restrict__ al_d,
                             const int* __restrict__ m_ord, const float* __restrict__ xp,
                             float* __restrict__ s_sum, float* __restrict__ agg, int E) {
  int e = blockIdx.x;
  if (e >= E) return;
  int t = threadIdx.x;
  int h = t >> 5;
  int lane = t & 31;
  int sN = esrc[e], dN = edst[e];
  float l = al_s[sN * HEADS + h] + al_d[dN * HEADS + h];
  l = l > 0.f ? l : 0.2f * l;
  float m = unord_f32(m_ord[dN * HEADS + h]);
  float ee = __expf(l - m);
  if (lane == 0) atomicAdd(&s_sum[dN * HEADS + h], ee);
  const float* xr = xp + (size_t)sN * (HEADS * DIM) + h * DIM;
  float* ar = agg + (size_t)dN * (HEADS * DIM) + h * DIM;
#pragma unroll
  for (int j = 0; j < 4; ++j) {
    int c = lane + j * 32;
    atomicAdd(&ar[c], ee * xr[c]);
  }
}

// ---------- GAT finalize: normalize, mean heads, add bias ----------
__global__ void gat_final(const float* __restrict__ agg, const float* __restrict__ s_sum,
                          const float* __restrict__ bg, float* __restrict__ hout, int Nrow) {
  int i = blockIdx.x * blockDim.x + threadIdx.x;
  if (i >= Nrow * DIM) return;
  int n = i >> 7, c = i & (DIM - 1);
  float acc = 0.f;
#pragma unroll
  for (int h = 0; h < HEADS; ++h)
    acc += agg[(size_t)n * (HEADS * DIM) + h * DIM + c] / (s_sum[n * HEADS + h] + 1e-16f);
  hout[i] = acc * (1.0f / HEADS) + bg[c];
}

// ---------- SAGE: scatter-add neighbor features + degree counts (wave per edge) ----------
__global__ void sage_scatter(const int* __restrict__ esrc, const int* __restrict__ edst,
                             const float* __restrict__ x, float* __restrict__ sums,
                             float* __restrict__ cnt, int E) {
  int e = blockIdx.x * (blockDim.x >> 5) + (threadIdx.x >> 5);
  if (e >= E) return;
  int lane = threadIdx.x & 31;
  int sN = esrc[e], dN = edst[e];
  const float* xr = x + (size_t)sN * DIM;
  float* sr = sums + (size_t)dN * DIM;
#pragma unroll
  for (int j = 0; j < 4; ++j) {
    int c = lane + j * 32;
    atomicAdd(&sr[c], xr[c]);
  }
  if (lane == 0) atomicAdd(&cnt[dN], 1.0f);
}

__global__ void sage_mean(float* __restrict__ sums, const float* __restrict__ cnt, int Nrow) {
  int i = blockIdx.x * blockDim.x + threadIdx.x;
  if (i >= Nrow * DIM) return;
  sums[i] /= fmaxf(cnt[i >> 7], 1.0f);
}

// ---------- residual + LayerNorm + ELU (wave32 per row; hB may be null) ----------
__global__ void post_ln_elu(const float* __restrict__ hA, const float* __restrict__ hB,
                            const float* __restrict__ xres, const float* __restrict__ g,
                            const float* __restrict__ b, float* __restrict__ out, int Nrow) {
  int row = blockIdx.x * (blockDim.x >> 5) + (threadIdx.x >> 5);
  if (row >= Nrow) return;
  int lane = threadIdx.x & 31;
  size_t base = (size_t)row * DIM;
  float v[4];
  float s = 0.f;
#pragma unroll
  for (int j = 0; j < 4; ++j) {
    int c = lane + j * 32;
    float t = hA[base + c] + xres[base + c];
    if (hB) t += hB[base + c];
    v[j] = t;
    s += t;
  }
#pragma unroll
  for (int o = 16; o > 0; o >>= 1) s += __shfl_xor(s, o, 32);
  float mu = s * (1.0f / DIM);
  float var = 0.f;
#pragma unroll
  for (int j = 0; j < 4; ++j) {
    float d2 = v[j] - mu;
    var += d2 * d2;
  }
#pragma unroll
  for (int o = 16; o > 0; o >>= 1) var += __shfl_xor(var, o, 32);
  var *= (1.0f / DIM);
  float inv = rsqrtf(var + 1e-5f);
#pragma unroll
  for (int j = 0; j < 4; ++j) {
    int c = lane + j * 32;
    float t = (v[j] - mu) * inv * g[c] + b[c];
    out[base + c] = t > 0.f ? t : expm1f(t);
  }
}

extern "C" void kernel_launch(void* const* d_in, const int* in_sizes, int n_in,
                              void* d_out, int out_size, void* d_ws, size_t ws_size,
                              hipStream_t stream) {
  (void)n_in; (void)out_size; (void)ws_size;
  const float* x_ent  = (const float*)d_in[0];
  const float* x_txn  = (const float*)d_in[1];
  const float* W_gat  = (const float*)d_in[2];
  const float* att_s  = (const float*)d_in[3];
  const float* att_d  = (const float*)d_in[4];
  const float* b_gat  = (const float*)d_in[5];
  const float* Wl_et  = (const float*)d_in[6];
  const float* bl_et  = (const float*)d_in[7];
  const float* Wr_et  = (const float*)d_in[8];
  const float* Wl_te  = (const float*)d_in[9];
  const float* bl_te  = (const float*)d_in[10];
  const float* Wr_te  = (const float*)d_in[11];
  const float* g_ent  = (const float*)d_in[12];
  const float* be_ent = (const float*)d_in[13];
  const float* g_txn  = (const float*)d_in[14];
  const float* be_txn = (const float*)d_in[15];
  const int* ee_src = (const int*)d_in[16];
  const int* ee_dst = (const int*)d_in[17];
  const int* et_src = (const int*)d_in[18];
  const int* et_dst = (const int*)d_in[19];
  const int* te_src = (const int*)d_in[20];
  const int* te_dst = (const int*)d_in[21];

  const int N_ENT = in_sizes[0] / DIM;
  const int N_TXN = in_sizes[1] / DIM;
  const int E_EE = in_sizes[16];
  const int E_ET = in_sizes[18];
  const int E_TE = in_sizes[20];

  // ---- workspace layout (floats) ----
  float* ws = (float*)d_ws;
  size_t off = 0;
  float* xp      = ws + off; off += (size_t)N_ENT * (HEADS * DIM);
  float* al_s    = ws + off; off += (size_t)N_ENT * HEADS;
  float* al_d    = ws + off; off += (size_t)N_ENT * HEADS;
  int*   m_ord   = (int*)(ws + off); off += (size_t)N_ENT * HEADS;
  size_t zoff = off;                       // zero-filled region start
  float* s_sum   = ws + off; off += (size_t)N_ENT * HEADS;
  float* agg     = ws + off; off += (size_t)N_ENT * (HEADS * DIM);
  float* sums_te = ws + off; off += (size_t)N_ENT * DIM;
  float* cnt_te  = ws + off; off += (size_t)N_ENT;
  float* sums_et = ws + off; off += (size_t)N_TXN * DIM;
  float* cnt_et  = ws + off; off += (size_t)N_TXN;
  size_t zcnt = off - zoff;
  float* h_gat   = ws + off; off += (size_t)N_ENT * DIM;
  float* h_te    = ws + off; off += (size_t)N_ENT * DIM;
  float* h_et    = ws + off; off += (size_t)N_TXN * DIM;

  float* out_ent = (float*)d_out;
  float* out_txn = out_ent + (size_t)N_ENT * DIM;

  const dim3 B256(256);
  const int ORD_NEG_INF = (int)0x807FFFFF;  // ord_f32(-inf)

  // init accumulators (fresh every call -> deterministic)
  fill_f32<<<dim3(2048), B256, 0, stream>>>(ws + zoff, 0.0f, zcnt);
  fill_i32<<<dim3(256), B256, 0, stream>>>(m_ord, ORD_NEG_INF, (size_t)N_ENT * HEADS);

  // GAT projection: xp[N,H*D] = x_ent @ W_gat   (M=50000, N=512, K=128)
  gemm_wmma_k128<<<dim3((HEADS * DIM) / 64, N_ENT / 16), dim3(32), 0, stream>>>(
      x_ent, W_gat, nullptr, nullptr, nullptr, xp, HEADS * DIM);

  att_logits<<<dim3((N_ENT * HEADS + 255) / 256), B256, 0, stream>>>(
      xp, att_s, att_d, al_s, al_d, N_ENT * HEADS);

  gat_edge_max<<<dim3((E_EE + 255) / 256), B256, 0, stream>>>(
      ee_src, ee_dst, al_s, al_d, m_ord, E_EE);

  gat_edge_acc<<<dim3(E_EE), dim3(128), 0, stream>>>(
      ee_src, ee_dst, al_s, al_d, m_ord, xp, s_sum, agg, E_EE);

  gat_final<<<dim3((N_ENT * DIM + 255) / 256), B256, 0, stream>>>(
      agg, s_sum, b_gat, h_gat, N_ENT);

  // SAGE txn->ent
  sage_scatter<<<dim3((E_TE + 7) / 8), B256, 0, stream>>>(
      te_src, te_dst, x_txn, sums_te, cnt_te, E_TE);
  sage_mean<<<dim3((N_ENT * DIM + 255) / 256), B256, 0, stream>>>(sums_te, cnt_te, N_ENT);
  gemm_wmma_k128<<<dim3(DIM / 64, N_ENT / 16), dim3(32), 0, stream>>>(
      sums_te, Wl_te, x_ent, Wr_te, bl_te, h_te, DIM);

  // SAGE ent->txn
  sage_scatter<<<dim3((E_ET + 7) / 8), B256, 0, stream>>>(
      et_src, et_dst, x_ent, sums_et, cnt_et, E_ET);
  sage_mean<<<dim3((N_TXN * DIM + 255) / 256), B256, 0, stream>>>(sums_et, cnt_et, N_TXN);
  gemm_wmma_k128<<<dim3(DIM / 64, N_TXN / 16), dim3(32), 0, stream>>>(
      sums_et, Wl_et, x_txn, Wr_et, bl_et, h_et, DIM);

  // residual + LayerNorm + ELU
  post_ln_elu<<<dim3((N_ENT + 7) / 8), B256, 0, stream>>>(
      h_gat, h_te, x_ent, g_ent, be_ent, out_ent, N_ENT);
  post_ln_elu<<<dim3((N_TXN + 7) / 8), B256, 0, stream>>>(
      h_et, nullptr, x_txn, g_txn, be_txn, out_txn, N_TXN);
}